// ModelNew_2911987826867
// MI455X (gfx1250) — compile-verified
//
#include <hip/hip_runtime.h>

typedef __attribute__((ext_vector_type(2))) float v2f;
typedef __attribute__((ext_vector_type(8))) float v8f;

#define WMMA_F32X4(a, b, c) \
  __builtin_amdgcn_wmma_f32_16x16x4_f32(false, (a), false, (b), (short)0, (c), false, false)

constexpr int Bn = 4, Sn = 4096, Hn = 16, Dn = 32, Cn = 16;

__device__ __forceinline__ size_t gidx(int b, int s, int h, int d) {
  return (((size_t)b * Sn + s) * Hn + h) * Dn + d;
}

// ---------------------------------------------------------------------------
// Pass 1: per (bh, segment): state contribution from zero init + gate product.
// One wave (32 lanes) per block.
// ---------------------------------------------------------------------------
__global__ __launch_bounds__(32)
void gdelta_pass1(const float* __restrict__ k, const float* __restrict__ v,
                  const float* __restrict__ g,
                  float* __restrict__ seg_state, float* __restrict__ seg_gprod,
                  int nseg, int L) {
  __shared__ float sK[16][33];   // K chunk (t, d)
  __shared__ float sVt[16][33];  // v~ = v / G  (t, m)
  __shared__ float sGc[32];      // per-chunk total gate product per m

  const int lane = threadIdx.x;
  const int hx = lane >> 4, l16 = lane & 15, koff = 2 * hx;
  const int seg = blockIdx.x % nseg, bh = blockIdx.x / nseg;
  const int b = bh / Hn, h = bh % Hn;
  const int s_base = seg * L;

  float stcol[32];  // not used; state kept in LDS below
  (void)stcol;
  __shared__ float sS[32][33];
  for (int d = 0; d < Dn; ++d) sS[d][lane] = 0.f;
  float segG = 1.f;
  __syncthreads();

  for (int c = 0; c < L; c += Cn) {
    const int s0 = s_base + c;
    // stage K chunk into LDS (each lane covers its koff pair of every group of 4)
    const float* krow = k + gidx(b, s0 + l16, h, 0);
#pragma unroll
    for (int i = 0; i < 8; ++i) {
      v2f kv = *(const v2f*)(krow + 4 * i + koff);
      sK[l16][4 * i + koff] = kv.x;
      sK[l16][4 * i + koff + 1] = kv.y;
    }
    // gates & v~: lane owns column m = lane
    float gl[16], vl[16];
#pragma unroll
    for (int t = 0; t < 16; ++t) {
      gl[t] = g[gidx(b, s0 + t, h, lane)];
      vl[t] = v[gidx(b, s0 + t, h, lane)];
    }
    float run = 1.f;
#pragma unroll
    for (int t = 0; t < 16; ++t) {
      run *= gl[t];
      sVt[t][lane] = vl[t] / fmaxf(run, 1e-30f);
    }
    sGc[lane] = run;
    segG *= run;
    __syncthreads();

    // state update tiles: S = Gc ⊙ (S + K^T v~)
#pragma unroll
    for (int dt = 0; dt < Dn; dt += 16) {
#pragma unroll
      for (int n0 = 0; n0 < Dn; n0 += 16) {
        v8f sacc = {};
#pragma unroll
        for (int si = 0; si < 4; ++si) {
          v2f a, bb;
          a.x = sK[4 * si + koff][dt + l16];      // A = K^T : row d, col s
          a.y = sK[4 * si + koff + 1][dt + l16];
          bb.x = sVt[4 * si + koff][n0 + l16];
          bb.y = sVt[4 * si + koff + 1][n0 + l16];
          sacc = WMMA_F32X4(a, bb, sacc);
        }
        const float gcm = sGc[n0 + l16];
#pragma unroll
        for (int vv = 0; vv < 8; ++vv) {
          const int d = dt + vv + 8 * hx, m = n0 + l16;
          sS[d][m] = gcm * (sS[d][m] + sacc[vv]);
        }
      }
    }
    __syncthreads();
  }

  float* sp = seg_state + ((size_t)bh * nseg + seg) * Dn * Dn;
  for (int d = 0; d < Dn; ++d) sp[d * Dn + lane] = sS[d][lane];
  seg_gprod[((size_t)bh * nseg + seg) * Dn + lane] = segG;
}

// ---------------------------------------------------------------------------
// Pass 2: per bh, sequential combine of segment transfer functions.
// carry' = Gseg ⊙ carry + seg_state; incoming state per segment stored.
// ---------------------------------------------------------------------------
__global__ __launch_bounds__(32)
void gdelta_pass2(const float* __restrict__ seg_state,
                  const float* __restrict__ seg_gprod,
                  float* __restrict__ seg_init, int nseg) {
  const int lane = threadIdx.x;
  const int bh = blockIdx.x;
  float carry[32];
#pragma unroll
  for (int d = 0; d < 32; ++d) carry[d] = 0.f;
  for (int sgi = 0; sgi < nseg; ++sgi) {
    float* ip = seg_init + ((size_t)bh * nseg + sgi) * Dn * Dn;
    const float* sp = seg_state + ((size_t)bh * nseg + sgi) * Dn * Dn;
    const float gp = seg_gprod[((size_t)bh * nseg + sgi) * Dn + lane];
#pragma unroll
    for (int d = 0; d < 32; ++d) {
      ip[d * Dn + lane] = carry[d];
      carry[d] = gp * carry[d] + sp[d * Dn + lane];
    }
  }
}

// ---------------------------------------------------------------------------
// Pass 3: per (bh, segment): chunked recurrence with true init, emit outputs.
//   A = causal(Q K^T); out = G ⊙ (A v~ + Q S0); S = Gc ⊙ (S + K^T v~)
// ---------------------------------------------------------------------------
__global__ __launch_bounds__(32)
void gdelta_pass3(const float* __restrict__ q, const float* __restrict__ k,
                  const float* __restrict__ v, const float* __restrict__ g,
                  const float* __restrict__ seg_init, float* __restrict__ out,
                  int nseg, int L) {
  __shared__ float sK[16][33];
  __shared__ float sVt[16][33];
  __shared__ float sG[16][33];
  __shared__ float sA[16][17];
  __shared__ float sS[32][33];

  const int lane = threadIdx.x;
  const int hx = lane >> 4, l16 = lane & 15, koff = 2 * hx;
  const int seg = blockIdx.x % nseg, bh = blockIdx.x / nseg;
  const int b = bh / Hn, h = bh % Hn;
  const int s_base = seg * L;

  if (seg_init) {
    const float* ip = seg_init + ((size_t)bh * nseg + seg) * Dn * Dn;
    for (int d = 0; d < Dn; ++d) sS[d][lane] = ip[d * Dn + lane];
  } else {
    for (int d = 0; d < Dn; ++d) sS[d][lane] = 0.f;
  }
  __syncthreads();

  for (int c = 0; c < L; c += Cn) {
    const int s0 = s_base + c;
    const float* qrow = q + gidx(b, s0 + l16, h, 0);
    const float* krow = k + gidx(b, s0 + l16, h, 0);
    // prefetch next chunk's rows while we compute this one
    if (c + Cn < L) {
      __builtin_prefetch(qrow + (size_t)Cn * Hn * Dn, 0, 0);
      __builtin_prefetch(krow + (size_t)Cn * Hn * Dn, 0, 0);
      __builtin_prefetch(v + gidx(b, s0 + Cn + l16, h, 0), 0, 0);
      __builtin_prefetch(g + gidx(b, s0 + Cn + l16, h, 0), 0, 0);
    }

    // Q/K fragments in WMMA A/B layout straight from global (float2 per group of 4)
    v2f qf[8], kf[8];
#pragma unroll
    for (int i = 0; i < 8; ++i) {
      qf[i] = *(const v2f*)(qrow + 4 * i + koff);
      kf[i] = *(const v2f*)(krow + 4 * i + koff);
    }
#pragma unroll
    for (int i = 0; i < 8; ++i) {
      sK[l16][4 * i + koff] = kf[i].x;
      sK[l16][4 * i + koff + 1] = kf[i].y;
    }

    // gates / v~: lane owns column m = lane
    float gl[16], vl[16];
#pragma unroll
    for (int t = 0; t < 16; ++t) {
      gl[t] = g[gidx(b, s0 + t, h, lane)];
      vl[t] = v[gidx(b, s0 + t, h, lane)];
    }
    float run = 1.f;
#pragma unroll
    for (int t = 0; t < 16; ++t) {
      run *= gl[t];
      sG[t][lane] = run;
      sVt[t][lane] = vl[t] / fmaxf(run, 1e-30f);
    }
    __syncthreads();

    // A = causal(Q K^T): 8 x wmma f32 16x16x4 over d
    v8f aacc = {};
#pragma unroll
    for (int i = 0; i < 8; ++i) aacc = WMMA_F32X4(qf[i], kf[i], aacc);
#pragma unroll
    for (int vv = 0; vv < 8; ++vv) {
      const int t = vv + 8 * hx;
      sA[t][l16] = (l16 <= t) ? aacc[vv] : 0.f;  // causal mask incl. diagonal
    }
    __syncthreads();

    // out = G ⊙ (A v~ + Q S0), two 16-wide n-tiles
#pragma unroll
    for (int n0 = 0; n0 < Dn; n0 += 16) {
      v8f oacc = {};
#pragma unroll
      for (int si = 0; si < 4; ++si) {
        v2f a, bb;
        a.x = sA[l16][4 * si + koff];
        a.y = sA[l16][4 * si + koff + 1];
        bb.x = sVt[4 * si + koff][n0 + l16];
        bb.y = sVt[4 * si + koff + 1][n0 + l16];
        oacc = WMMA_F32X4(a, bb, oacc);
      }
#pragma unroll
      for (int i = 0; i < 8; ++i) {
        v2f bb;
        bb.x = sS[4 * i + koff][n0 + l16];
        bb.y = sS[4 * i + koff + 1][n0 + l16];
        oacc = WMMA_F32X4(qf[i], bb, oacc);
      }
#pragma unroll
      for (int vv = 0; vv < 8; ++vv) {
        const int t = vv + 8 * hx, m = n0 + l16;
        out[gidx(b, s0 + t, h, m)] = oacc[vv] * sG[t][m];
      }
    }
    __syncthreads();

    // S = Gc ⊙ (S + K^T v~)
#pragma unroll
    for (int dt = 0; dt < Dn; dt += 16) {
#pragma unroll
      for (int n0 = 0; n0 < Dn; n0 += 16) {
        v8f sacc = {};
#pragma unroll
        for (int si = 0; si < 4; ++si) {
          v2f a, bb;
          a.x = sK[4 * si + koff][dt + l16];
          a.y = sK[4 * si + koff + 1][dt + l16];
          bb.x = sVt[4 * si + koff][n0 + l16];
          bb.y = sVt[4 * si + koff + 1][n0 + l16];
          sacc = WMMA_F32X4(a, bb, sacc);
        }
        const float gcm = sG[15][n0 + l16];
#pragma unroll
        for (int vv = 0; vv < 8; ++vv) {
          const int d = dt + vv + 8 * hx, m = n0 + l16;
          sS[d][m] = gcm * (sS[d][m] + sacc[vv]);
        }
      }
    }
    __syncthreads();
  }
}

extern "C" void kernel_launch(void* const* d_in, const int* in_sizes, int n_in,
                              void* d_out, int out_size, void* d_ws, size_t ws_size,
                              hipStream_t stream) {
  (void)in_sizes; (void)n_in; (void)out_size;
  const float* q = (const float*)d_in[0];
  const float* k = (const float*)d_in[1];
  const float* v = (const float*)d_in[2];
  const float* g = (const float*)d_in[3];
  float* out = (float*)d_out;

  const int BH = Bn * Hn;
  const int nseg = 32;                 // 2048-way parallel scan
  const int L = Sn / nseg;             // 128 steps per segment
  const size_t st_elems = (size_t)BH * nseg * Dn * Dn;
  const size_t gp_elems = (size_t)BH * nseg * Dn;
  const size_t need = (2 * st_elems + gp_elems) * sizeof(float);

  if (d_ws && ws_size >= need) {
    float* seg_state = (float*)d_ws;
    float* seg_gprod = seg_state + st_elems;
    float* seg_init = seg_gprod + gp_elems;
    gdelta_pass1<<<BH * nseg, 32, 0, stream>>>(k, v, g, seg_state, seg_gprod, nseg, L);
    gdelta_pass2<<<BH, 32, 0, stream>>>(seg_state, seg_gprod, seg_init, nseg);
    gdelta_pass3<<<BH * nseg, 32, 0, stream>>>(q, k, v, g, seg_init, out, nseg, L);
  } else {
    // fallback: fully serial per head, zero init, single pass
    gdelta_pass3<<<BH, 32, 0, stream>>>(q, k, v, g, nullptr, out, 1, Sn);
  }
}